// MetaNet_13666585936249
// MI455X (gfx1250) — compile-verified
//
#include <hip/hip_runtime.h>
#include <hip/hip_bf16.h>

// ---- problem constants (from reference) ----
#define N_LAYERS 14
#define HID      128
#define FEAT     76128              // 2080 + 8256 + 32896 + 32896
#define HROWS    (128 * N_LAYERS)   // 1792
#define OUTP     36864
#define NQUADS   (FEAT / 4)         // 19032 K-quads for stage 3
#define KSPLIT   48                 // K-chunks for stage-3 occupancy
#define QPC      398                // quads/chunk: even, 48*398 >= 19032

typedef __attribute__((ext_vector_type(2))) float v2f;
typedef __attribute__((ext_vector_type(4))) float v4f;
typedef __attribute__((ext_vector_type(8))) float v8f;

// ---- workspace layout (floats) ----
#define WS_G4P     0          // 65536
#define WS_FEAT    65536      // 76128
#define WS_PART    141664     // KSPLIT*HROWS = 86016
#define WS_HID     227680     // 1792

// ---------------------------------------------------------------- avg pool
__global__ void k_avgpool(const float* __restrict__ g4, float* __restrict__ g4p) {
    int i = blockIdx.x * blockDim.x + threadIdx.x;     // 65536 outputs
    int r = i >> 8, c = i & 255;
    const float* p = g4 + (size_t)(r * 2) * 512 + c * 2;
    g4p[i] = 0.25f * (p[0] + p[1] + p[512] + p[513]);
}

// ------------------------------------------------- triu gather (row-major)
__global__ void k_triu(const float* __restrict__ src, int C, int base,
                       float* __restrict__ feat) {
    int i = blockIdx.x * blockDim.x + threadIdx.x;
    if (i >= C * C) return;
    int r = i / C, c = i % C;
    if (c < r) return;
    int off = r * C - (r * (r - 1)) / 2 + (c - r);     // elems before row r + col
    feat[base + off] = src[i];
}

// ---------------------------------------- stage 3: hidden GEMV (K-split, WMMA)
// wave = (row-tile, k-chunk). A = 16 rows x 4 K of Wh. B col 0 = feat quad.
// B layout (mirrors documented A layout): v0 = K{0 lo-half, 2 hi-half},
// v1 = K{1,3}; only column N=0 (lanes 0 and 16) is nonzero.
__global__ void k_gemv_hidden(const float* __restrict__ Wh,
                              const float* __restrict__ feat,
                              float* __restrict__ partial) {
    const int lane = threadIdx.x & 31;
    // wave-uniform scalars: scalar loop control + uniform (SMEM) feat loads
    const int wave = __builtin_amdgcn_readfirstlane(
        (blockIdx.x * blockDim.x + threadIdx.x) >> 5);
    const int tiles = HROWS / 16;              // 112
    const int rt = wave % tiles;
    const int s  = wave / tiles;               // k-chunk id
    const int rb = rt * 16;

    int q0 = s * QPC;
    int q1 = q0 + QPC; if (q1 > NQUADS) q1 = NQUADS;   // count always even

    const int m    = lane & 15;
    const int koff = (lane >> 4) ? 2 : 0;      // lo half-wave K{0,1}, hi K{2,3}
    const float* rowp = Wh + (size_t)(rb + m) * FEAT + koff;

    const bool l0  = (lane == 0);
    const bool l16 = (lane == 16);

    v8f acc0 = {}, acc1 = {};
    for (int q = q0; q < q1; q += 2) {         // 2 quads/iter: 2x MLP
        v2f a0 = *(const v2f*)(rowp + 4 * q);
        v2f a1 = *(const v2f*)(rowp + 4 * q + 4);
        __builtin_prefetch((const void*)(rowp + 4 * q + 128), 0, 1);
        v4f f0 = *(const v4f*)(feat + 4 * q);          // uniform -> s_load
        v4f f1 = *(const v4f*)(feat + 4 * q + 4);
        v2f b0, b1;
        b0.x = l0 ? f0.x : (l16 ? f0.z : 0.f);
        b0.y = l0 ? f0.y : (l16 ? f0.w : 0.f);
        b1.x = l0 ? f1.x : (l16 ? f1.z : 0.f);
        b1.y = l0 ? f1.y : (l16 ? f1.w : 0.f);
        acc0 = __builtin_amdgcn_wmma_f32_16x16x4_f32(
            false, a0, false, b0, (short)0, acc0, false, false);
        acc1 = __builtin_amdgcn_wmma_f32_16x16x4_f32(
            false, a1, false, b1, (short)0, acc1, false, false);
    }
    v8f acc = acc0 + acc1;
    // D column N=0: lane 0 holds M=0..7, lane 16 holds M=8..15
    if (lane == 0) {
        #pragma unroll
        for (int r = 0; r < 8; ++r) partial[(size_t)s * HROWS + rb + r] = acc[r];
    } else if (lane == 16) {
        #pragma unroll
        for (int r = 0; r < 8; ++r) partial[(size_t)s * HROWS + rb + 8 + r] = acc[r];
    }
}

// ------------------------------------------------ bias + relu + chunk reduce
__global__ void k_reduce_relu(const float* __restrict__ partial,
                              const float* __restrict__ bh,
                              float* __restrict__ hidden) {
    int r = blockIdx.x * blockDim.x + threadIdx.x;
    if (r >= HROWS) return;
    float s = bh[r];
    #pragma unroll 4
    for (int i = 0; i < KSPLIT; ++i) s += partial[(size_t)i * HROWS + r];
    hidden[r] = s > 0.f ? s : 0.f;
}

// --------------------------------- stage 4: 14 per-head GEMVs (K=128, WMMA)
__global__ void k_heads(const float* __restrict__ Wf,
                        const float* __restrict__ bf,
                        const float* __restrict__ hidden,
                        float* __restrict__ out) {
    const int lane = threadIdx.x & 31;
    const int wave = __builtin_amdgcn_readfirstlane(
        (blockIdx.x * blockDim.x + threadIdx.x) >> 5);
    const int tilesPerHead = OUTP / 16;        // 2304
    const int n  = wave / tilesPerHead;
    const int pb = (wave % tilesPerHead) * 16;

    const int m    = lane & 15;
    const int koff = (lane >> 4) ? 2 : 0;
    const float* rowp = Wf + ((size_t)n * OUTP + pb + m) * HID + koff;
    const float* h    = hidden + n * HID;      // uniform -> s_load broadcasts

    const bool l0  = (lane == 0);
    const bool l16 = (lane == 16);

    v8f acc0 = {}, acc1 = {};
    #pragma unroll
    for (int q = 0; q < HID / 4; q += 2) {     // 32 WMMAs, 2 acc chains
        v2f a0 = *(const v2f*)(rowp + 4 * q);
        v2f a1 = *(const v2f*)(rowp + 4 * q + 4);
        v4f f0 = *(const v4f*)(h + 4 * q);
        v4f f1 = *(const v4f*)(h + 4 * q + 4);
        v2f b0, b1;
        b0.x = l0 ? f0.x : (l16 ? f0.z : 0.f);
        b0.y = l0 ? f0.y : (l16 ? f0.w : 0.f);
        b1.x = l0 ? f1.x : (l16 ? f1.z : 0.f);
        b1.y = l0 ? f1.y : (l16 ? f1.w : 0.f);
        acc0 = __builtin_amdgcn_wmma_f32_16x16x4_f32(
            false, a0, false, b0, (short)0, acc0, false, false);
        acc1 = __builtin_amdgcn_wmma_f32_16x16x4_f32(
            false, a1, false, b1, (short)0, acc1, false, false);
    }
    v8f acc = acc0 + acc1;
    size_t ob = (size_t)n * OUTP + pb;
    if (lane == 0) {
        #pragma unroll
        for (int r = 0; r < 8; ++r) out[ob + r] = acc[r] + bf[ob + r];
    } else if (lane == 16) {
        #pragma unroll
        for (int r = 0; r < 8; ++r) out[ob + 8 + r] = acc[r] + bf[ob + 8 + r];
    }
}

extern "C" void kernel_launch(void* const* d_in, const int* in_sizes, int n_in,
                              void* d_out, int out_size, void* d_ws, size_t ws_size,
                              hipStream_t stream) {
    const float* g1 = (const float*)d_in[0];
    const float* g2 = (const float*)d_in[1];
    const float* g3 = (const float*)d_in[2];
    const float* g4 = (const float*)d_in[3];
    const float* Wh = (const float*)d_in[4];
    const float* bh = (const float*)d_in[5];
    const float* Wf = (const float*)d_in[6];
    const float* bf = (const float*)d_in[7];
    float* out = (float*)d_out;

    float* ws      = (float*)d_ws;
    float* g4p     = ws + WS_G4P;
    float* feat    = ws + WS_FEAT;
    float* partial = ws + WS_PART;
    float* hidden  = ws + WS_HID;

    // 1) avg-pool g4 -> g4p  (65536 outputs)
    k_avgpool<<<256, 256, 0, stream>>>(g4, g4p);

    // 2) triu gathers into feat (bases: 0, 2080, 10336, 43232)
    k_triu<<<(64 * 64 + 255) / 256,  256, 0, stream>>>(g1,  64,     0, feat);
    k_triu<<<(128 * 128 + 255) / 256, 256, 0, stream>>>(g2, 128,  2080, feat);
    k_triu<<<(256 * 256 + 255) / 256, 256, 0, stream>>>(g3, 256, 10336, feat);
    k_triu<<<(256 * 256 + 255) / 256, 256, 0, stream>>>(g4p, 256, 43232, feat);

    // 3) hidden GEMV: 112 row-tiles x 48 k-chunks = 5376 waves (8 waves/block)
    k_gemv_hidden<<<(112 * KSPLIT) / 8, 256, 0, stream>>>(Wh, feat, partial);

    // 4) reduce partials + bias + relu
    k_reduce_relu<<<(HROWS + 255) / 256, 256, 0, stream>>>(partial, bh, hidden);

    // 5) per-head GEMVs: 14 * 2304 = 32256 waves (8 waves/block)
    k_heads<<<(N_LAYERS * (OUTP / 16)) / 8, 256, 0, stream>>>(Wf, bf, hidden, out);
}